// PDELayer_20718922236466
// MI455X (gfx1250) — compile-verified
//
#include <hip/hip_runtime.h>
#include <stdint.h>

// ---------------------------------------------------------------------------
// 2-D variable-coefficient heat equation, 10 fused Euler steps in LDS.
// One workgroup (256 threads) per batch image (48x48 f32).
// Inbound AND outbound tiles via CDNA5 Tensor Data Mover (TENSORcnt),
// with plain coalesced-load/store fallbacks if the builtins are unavailable.
// Memory-bound: ~302 MB total HBM traffic -> ~13 us floor at 23.3 TB/s.
// ---------------------------------------------------------------------------

constexpr int NG      = 48;          // grid edge
constexpr int NCELL   = NG * NG;     // 2304
constexpr int THREADS = 256;         // 8 wave32
constexpr int PER_THR = NCELL / THREADS; // 9
constexpr int NSTEPS  = 10;          // T/dt

typedef unsigned int u32x4 __attribute__((ext_vector_type(4)));
typedef int          i32x4 __attribute__((ext_vector_type(4)));
typedef int          i32x8 __attribute__((ext_vector_type(8)));

#if defined(__gfx1250__) && __has_builtin(__builtin_amdgcn_tensor_load_to_lds) && \
    __has_builtin(__builtin_amdgcn_s_wait_tensorcnt)
#define USE_TDM_LOAD 1
#endif
#if defined(__gfx1250__) && __has_builtin(__builtin_amdgcn_tensor_store_from_lds) && \
    __has_builtin(__builtin_amdgcn_s_wait_tensorcnt)
#define USE_TDM_STORE 1
#endif

#if defined(USE_TDM_LOAD) || defined(USE_TDM_STORE)
// D# for a contiguous 48x48 f32 2-D tile (groups 2/3 zero => up-to-2D form).
__device__ __forceinline__ void tdm_desc_48x48_f32(unsigned long long ga,
                                                   unsigned int lds,
                                                   u32x4& g0, i32x8& g1)
{
    g0[0] = 1u;                                    // count=1, user mode
    g0[1] = lds;                                   // lds_addr @ bits 63:32
    g0[2] = (unsigned int)ga;                      // global_addr lo
    g0[3] = ((unsigned int)(ga >> 32) & 0x01FFFFFFu) | 0x80000000u; // addr[56:32] | type=2

    g1[0] = 0x00020000;                            // workgroup_mask=0 (no cluster), data_size=4B
    g1[1] = (NG & 0xFFFF) << 16;                   // tensor_dim0[15:0] @ bits 63:48
    g1[2] = (NG & 0xFFFF) << 16;                   // tensor_dim1[15:0] @ bits 95:80
    g1[3] = (NG & 0xFFFF) << 16;                   // tile_dim0        @ bits 127:112
    g1[4] = NG;                                    // tile_dim1 (tile_dim2 = 0)
    g1[5] = NG;                                    // tensor_dim0_stride lo32
    g1[6] = (NCELL & 0xFFFF) << 16;                // tensor_dim1_stride lo16 (unused for 2D)
    g1[7] = 0;
}
#endif

__global__ __launch_bounds__(THREADS) void pde48_heat_kernel(
    const float* __restrict__ u0,
    const float* __restrict__ alpha_w,
    const float* __restrict__ beta_w,
    float* __restrict__ out)
{
    __shared__ float bufA[NCELL];
    __shared__ float bufB[NCELL];
    __shared__ float ringT[NG];   // frozen reflect pad: row -1   = init row 1
    __shared__ float ringD[NG];   //                     row  48  = init row 46
    __shared__ float ringL[NG];   //                     col -1   = init col 1
    __shared__ float ringR[NG];   //                     col  48  = init col 46
    __shared__ float alph[NG];    // 0.5*dt/dx^2 * (a0 + a1 sin(2pi y) + a2 sin(4pi y))
    __shared__ float beta[NG];    //     dt/dy^2 * (b0 + b1 cos(2pi x) + b2 cos(4pi x))

    const int tid = threadIdx.x;
    const float* __restrict__ g = u0 + (size_t)blockIdx.x * NCELL;

#ifdef USE_TDM_LOAD
    // ---- CDNA5 TDM async DMA: global 48x48 f32 tile -> LDS bufA ----
    if (tid < 32) {  // issue from wave 0 only (EXEC ignored; one issue per wave)
        u32x4 g0; i32x8 g1;
        tdm_desc_48x48_f32((unsigned long long)(uintptr_t)g,
                           (unsigned int)(uintptr_t)(&bufA[0]), g0, g1);
        i32x4 gz4 = {0, 0, 0, 0};
        i32x8 gz8 = {0, 0, 0, 0, 0, 0, 0, 0};
        __builtin_amdgcn_tensor_load_to_lds(g0, g1, gz4, gz4, gz8, 0);
        __builtin_amdgcn_s_wait_tensorcnt(0);
    }
    __syncthreads();
    asm volatile("" ::: "memory");  // TDM wrote LDS behind the compiler's back
#else
    // ---- fallback: coalesced loads into LDS ----
    #pragma unroll
    for (int k = 0; k < PER_THR; ++k) {
        const int idx = tid + k * THREADS;
        bufA[idx] = g[idx];
    }
    __syncthreads();
#endif

    // ---- frozen reflect ring + coefficient vectors (row-only / col-only) ----
    if (tid < NG) {
        const int i = tid;
        ringT[i] = bufA[1 * NG + i];
        ringD[i] = bufA[(NG - 2) * NG + i];
        ringL[i] = bufA[i * NG + 1];
        ringR[i] = bufA[i * NG + (NG - 2)];

        constexpr float TWO_PI = 6.2831853071795864769f;
        constexpr float ASC = (float)(0.5 * 0.001 / ((1.0 / 48.0) * (1.0 / 48.0))); // 1.152
        constexpr float BSC = (float)(1.0 * 0.001 / ((1.0 / 48.0) * (1.0 / 48.0))); // 2.304
        const float t = (float)i * (1.0f / 47.0f);     // linspace(0,1,48)

        alph[i] = ASC * (alpha_w[0] + alpha_w[1] * sinf(TWO_PI * t)
                                    + alpha_w[2] * sinf(2.0f * TWO_PI * t));
        beta[i] = BSC * (beta_w[0]  + beta_w[1]  * cosf(TWO_PI * t)
                                    + beta_w[2]  * cosf(2.0f * TWO_PI * t));
    }
    __syncthreads();

    // ---- per-thread cell assignment (9 cells each, coalesced layout) ----
    int rr[PER_THR], cc[PER_THR];
    #pragma unroll
    for (int k = 0; k < PER_THR; ++k) {
        const int idx = tid + k * THREADS;
        rr[k] = idx / NG;
        cc[k] = idx - rr[k] * NG;
    }

    // ---- 10 ping-pong Euler steps entirely in LDS ----
    float* cur = bufA;
    float* nxt = bufB;
    for (int t = 0; t < NSTEPS; ++t) {
        #pragma unroll
        for (int k = 0; k < PER_THR; ++k) {
            const int idx = tid + k * THREADS;
            const int r = rr[k], c = cc[k];
            const float ctr = cur[idx];
            const float up = (r == 0)      ? ringT[c] : cur[idx - NG];
            const float dn = (r == NG - 1) ? ringD[c] : cur[idx + NG];
            const float lf = (c == 0)      ? ringL[r] : cur[idx - 1];
            const float rt = (c == NG - 1) ? ringR[r] : cur[idx + 1];
            nxt[idx] = ctr + alph[r] * (up + dn - 2.0f * ctr)
                           + beta[c] * (lf + rt - 2.0f * ctr);
        }
        __syncthreads();   // also orders the final buffer before the TDM store
        float* tmp = cur; cur = nxt; nxt = tmp;
    }

    // ---- outbound: LDS tile -> global via TDM (or coalesced stores) ----
    float* __restrict__ o = out + (size_t)blockIdx.x * NCELL;
#ifdef USE_TDM_STORE
    if (tid < 32) {
        u32x4 g0; i32x8 g1;
        tdm_desc_48x48_f32((unsigned long long)(uintptr_t)o,
                           (unsigned int)(uintptr_t)cur, g0, g1);
        i32x4 gz4 = {0, 0, 0, 0};
        i32x8 gz8 = {0, 0, 0, 0, 0, 0, 0, 0};
        __builtin_amdgcn_tensor_store_from_lds(g0, g1, gz4, gz4, gz8, 0);
        __builtin_amdgcn_s_wait_tensorcnt(0);  // data committed before wave exit
    }
#else
    #pragma unroll
    for (int k = 0; k < PER_THR; ++k) {
        const int idx = tid + k * THREADS;
        o[idx] = cur[idx];
    }
#endif
}

extern "C" void kernel_launch(void* const* d_in, const int* in_sizes, int n_in,
                              void* d_out, int out_size, void* d_ws, size_t ws_size,
                              hipStream_t stream) {
    const float* u0 = (const float*)d_in[0];   // (B,1,48,48) f32
    const float* aw = (const float*)d_in[1];   // (3,) f32
    const float* bw = (const float*)d_in[2];   // (3,) f32
    float*      out = (float*)d_out;           // (B,1,48,48) f32

    const int B = in_sizes[0] / NCELL;         // 16384
    pde48_heat_kernel<<<dim3(B), dim3(THREADS), 0, stream>>>(u0, aw, bw, out);
}